// TypedBinaryTreeLSTMLayer_41497974014350
// MI455X (gfx1250) — compile-verified
//
#include <hip/hip_runtime.h>

// Problem constants (match reference)
#define B_    256
#define N_    16
#define M_    128
#define V_    64
#define H_    256
#define K_    8
#define NP1   17           // N+1
#define KO    136          // K*(N+1)
#define TYPES_ 12
#define MV    8192         // M*V
#define NEGV  (-1e30f)

typedef __attribute__((ext_vector_type(2))) float v2f;
typedef __attribute__((ext_vector_type(8))) float v8f;

// ---------------------------------------------------------------------------
// Kernel 0: zero the whole output (new_d tail + new_v must be zeros).
// ---------------------------------------------------------------------------
__global__ __launch_bounds__(256) void zero_kernel(float* __restrict__ p, int n) {
  int i  = blockIdx.x * blockDim.x + threadIdx.x;
  int i4 = i * 4;
  if (i4 + 3 < n) {
    *reinterpret_cast<float4*>(p + i4) = make_float4(0.f, 0.f, 0.f, 0.f);
  } else {
    for (int j = i4; j < n; ++j) p[j] = 0.f;
  }
}

// ---------------------------------------------------------------------------
// Kernel 1: per-batch gathered GEMV + gumbel-softmax straight-through -> dt
// One block per batch item, 256 threads.
// ---------------------------------------------------------------------------
__global__ __launch_bounds__(256) void dt_kernel(
    const float* __restrict__ type_emb,   // [21, H]
    const float* __restrict__ W_sem,      // [15, 12, H, KO]
    const float* __restrict__ b_sem,      // [15, 12, KO]
    const float* __restrict__ gumbel,     // [B, K, NP1]
    const int*   __restrict__ types,      // [B]
    const int*   __restrict__ spans,      // [B]
    float*       __restrict__ dt_out)     // [B, KO]
{
  __shared__ float te[H_];
  __shared__ float logits[KO];

  const int b   = blockIdx.x;
  const int tid = threadIdx.x;
  const int tt  = types[b];
  const int sp  = spans[b];

  te[tid] = type_emb[tt * H_ + tid];            // 256 threads == H_
  __syncthreads();

  const size_t gidx = (size_t)((sp - 2) * TYPES_ + (tt - 9));
  const float* Wg = W_sem + gidx * (size_t)(H_ * KO);
  const float* bg = b_sem + gidx * (size_t)KO;

  if (tid < KO) {
    float acc = bg[tid];
    for (int h = 0; h < H_; ++h)                // coalesced across tid per h
      acc += te[h] * Wg[(size_t)h * KO + tid];
    logits[tid] = acc;
  }
  __syncthreads();

  if (tid < K_) {
    const int k   = tid;
    const int lim = sp + 1;                     // valid: n < span+1
    float l[NP1], y[NP1];

    for (int n = 0; n < NP1; ++n)
      l[n] = (n < lim) ? logits[k * NP1 + n] : NEGV;

    // log_softmax
    float mx = l[0];
    for (int n = 1; n < NP1; ++n) mx = fmaxf(mx, l[n]);
    float s = 0.f;
    for (int n = 0; n < NP1; ++n) s += expf(l[n] - mx);
    const float lse = logf(s) + mx;

    // y_soft = softmax(logp + gumbel)   (TAU == 1)
    float z[NP1];
    float mx2 = -3.402823e38f;
    for (int n = 0; n < NP1; ++n) {
      z[n] = (l[n] - lse) + gumbel[b * KO + k * NP1 + n];
      mx2 = fmaxf(mx2, z[n]);
    }
    float den = 0.f;
    for (int n = 0; n < NP1; ++n) { y[n] = expf(z[n] - mx2); den += y[n]; }
    const float rden = 1.f / den;
    for (int n = 0; n < NP1; ++n) y[n] *= rden;

    // first-occurrence argmax (jnp.argmax semantics)
    int amax = 0; float best = y[0];
    for (int n = 1; n < NP1; ++n) if (y[n] > best) { best = y[n]; amax = n; }

    for (int n = 0; n < NP1; ++n) {
      const float hard = (n == amax) ? 1.f : 0.f;
      float v = (hard + y[n]) - y[n];           // straight-through fp value
      v = (n < lim) ? v : 0.f;                  // mask beyond span+1
      if (tt == 20)                             // fixed start template
        v = (k == 0) ? ((n == 1) ? 1.f : 0.f) : ((n == 0) ? 1.f : 0.f);
      dt_out[b * KO + k * NP1 + n] = v;
    }
  }
}

// choices[b][c][f]: row 0 = pad one-hot template, rows 1..16 = decodings,
// rows 17..19 = zero padding of the contraction dim.
__device__ __forceinline__ float choice_val(const float* __restrict__ decb,
                                            int c, int f) {
  if (c == 0)  return ((f & 63) == 0) ? 1.f : 0.f;
  if (c > 16)  return 0.f;
  return decb[(size_t)(c - 1) * MV + f];
}

// ---------------------------------------------------------------------------
// Kernel 2: outputs = dt[8x17(pad 16x20)] x choices[17x8192] via
// V_WMMA_F32_16X16X4_F32, staged entirely in LDS (256 KB), then fused
// trim-lengths + sequential concat written straight to new_d.
// One block per batch item, 8 waves.
// ---------------------------------------------------------------------------
__global__ __launch_bounds__(256) void apply_kernel(
    const float* __restrict__ dec,        // [B, N, M, V]
    const float* __restrict__ dt_g,       // [B, KO]
    float*       __restrict__ out)        // [B, M, V] region of d_out
{
  extern __shared__ float lout[];         // K_ * MV floats = 256 KB
  __shared__ float dt_s[KO];
  __shared__ int olen[K_], startk[K_], lenk[K_];

  const int b    = blockIdx.x;
  const int tid  = threadIdx.x;
  const int lane = tid & 31;
  const int wave = tid >> 5;

  if (tid < KO) dt_s[tid] = dt_g[b * KO + tid];
  if (tid < K_) olen[tid] = 0;
  __syncthreads();

  // A fragment (16x4 f32): lanes 0-15 hold K=0,1 ; lanes 16-31 hold K=2,3.
  const int mrow = lane & 15;
  const int koff = (lane < 16) ? 0 : 2;
  v2f aF[5];
#pragma unroll
  for (int s = 0; s < 5; ++s) {
    const int n0 = 4 * s + koff;
    aF[s].x = (mrow < K_ && n0     < NP1) ? dt_s[mrow * NP1 + n0]     : 0.f;
    aF[s].y = (mrow < K_ && n0 + 1 < NP1) ? dt_s[mrow * NP1 + n0 + 1] : 0.f;
  }

  const float* decb = dec + (size_t)b * (N_ * MV);

  // 512 N-tiles of 16 columns, round-robin over 8 waves. Uniform trip count
  // per wave -> EXEC all-ones at every WMMA issue.
  for (int t = wave; t < MV / 16; t += 8) {
    const int F = t * 16;
    const int f = F + mrow;                       // this lane's column
    v8f c = {};
#pragma unroll
    for (int s = 0; s < 5; ++s) {
      const int c0 = 4 * s + koff;                // B rows mirror A's K split
      v2f bF;
      bF.x = choice_val(decb, c0,     f);
      bF.y = choice_val(decb, c0 + 1, f);
      c = __builtin_amdgcn_wmma_f32_16x16x4_f32(
              false, aF[s], false, bF, (short)0, c, false, false);
    }
    // D rows 0..7 (lanes 0-15) are k=0..7; rows 8..15 are padding.
    if (lane < 16) {
#pragma unroll
      for (int r = 0; r < K_; ++r) lout[r * MV + f] = c[r];
    }
  }
  __syncthreads();

  // Per-k trimmed length: last row m whose argmax over V is non-zero.
  // argmax!=0  <=>  exists v>=1 with out[v] > out[0] (first-max semantics).
  for (int p = tid; p < K_ * M_; p += 256) {
    const int k = p >> 7;
    const int m = p & (M_ - 1);
    const float* row = lout + k * MV + m * V_;
    const float v0 = row[0];
    int flag = 0;
    for (int v = 1; v < V_; ++v)
      if (row[v] > v0) { flag = 1; break; }
    if (flag) atomicMax(&olen[k], m + 1);         // ds_max_rtn on LDS
  }
  __syncthreads();

  if (tid == 0) {
    int cur = 0;
    for (int k = 0; k < K_; ++k) {
      int ol = olen[k];
      if (ol > M_ - cur) ol = M_ - cur;
      startk[k] = cur; lenk[k] = ol; cur += ol;
    }
  }
  __syncthreads();

  // Sequential concat: result[start_k + r, :] = outputs[k, r, :]
  float* outb = out + (size_t)b * MV;
  for (int k = 0; k < K_; ++k) {
    const int cnt  = lenk[k] * V_;
    const int base = startk[k] * V_;
    for (int i = tid; i < cnt; i += 256)
      outb[base + i] = lout[k * MV + i];
  }
}

// ---------------------------------------------------------------------------
extern "C" void kernel_launch(void* const* d_in, const int* in_sizes, int n_in,
                              void* d_out, int out_size, void* d_ws, size_t ws_size,
                              hipStream_t stream) {
  const float* dec   = (const float*)d_in[0];   // decodings [B,N,M,V] f32
  // d_in[1] (variables) unused by the reference output
  const int*   types = (const int*)  d_in[2];   // target_types [B] i32
  const int*   spans = (const int*)  d_in[3];   // spans [B] i32
  const float* temb  = (const float*)d_in[4];   // type_emb_table [21,H]
  const float* Wsem  = (const float*)d_in[5];   // W_sem [15,12,H,136]
  const float* bsem  = (const float*)d_in[6];   // b_sem [15,12,136]
  const float* gumb  = (const float*)d_in[7];   // gumbel [B,K,17]
  float* out = (float*)d_out;                   // new_d ++ new_v (f32)
  float* dtw = (float*)d_ws;                    // dt scratch: B*136 floats

  const int nzBlocks = (out_size + 1023) / 1024;          // 4 floats/thread
  zero_kernel<<<nzBlocks, 256, 0, stream>>>(out, out_size);

  dt_kernel<<<B_, 256, 0, stream>>>(temb, Wsem, bsem, gumb, types, spans, dtw);

  const size_t shbytes = (size_t)K_ * MV * sizeof(float); // 256 KB LDS
  apply_kernel<<<B_, 256, shbytes, stream>>>(dec, dtw, out);
}